// SeparationParamSimpleLatent_24584392802959
// MI455X (gfx1250) — compile-verified
//
#include <hip/hip_runtime.h>
#include <hip/hip_bf16.h>

// ---------------------------------------------------------------------------
// SeparationParamSimpleLatent — CDNA5 (gfx1250) implementation.
//
// Exact algebraic folding: h is a broadcast of h0 over (B,T,X), so
// sin(h@W1+b1) is one 128-vector s; the BlockMod+readout reduces to
//   u[b,t,x] = (1/32) * sum_k zt[b,t,k] * px[b,x,k] * v[k>>5] + c
// with v[d] = sum_j Wc[d,j]*s[j]*(W2@dW)[j] and scalar c.
// All heavy GEMMs run on V_WMMA_F32_16X16X4_F32 (fp32 wave32 WMMA).
// ---------------------------------------------------------------------------

typedef __attribute__((ext_vector_type(2))) float v2f;
typedef __attribute__((ext_vector_type(8))) float v8f;

#define HIDDEN 128
#define EMB    128
#define DHL    1024   // DH*LATENT
#define B_     4
#define X_     128
#define T_     256

// ---------------- Stage 0: tiny param-side MLPs + folded constants ---------
__global__ __launch_bounds__(128)
void setup_kernel(const float* __restrict__ param,
                  const float* __restrict__ p2e_W1, const float* __restrict__ p2e_b1,
                  const float* __restrict__ p2e_W2, const float* __restrict__ p2e_b2,
                  const float* __restrict__ e2ex_W, const float* __restrict__ e2ex_b,
                  const float* __restrict__ e2et_W, const float* __restrict__ e2et_b,
                  const float* __restrict__ px_We,  const float* __restrict__ px_be,
                  const float* __restrict__ zt_We,  const float* __restrict__ zt_be,
                  const float* __restrict__ h0,
                  const float* __restrict__ blk_W1, const float* __restrict__ blk_b1,
                  const float* __restrict__ blk_Wc, const float* __restrict__ blk_bc,
                  const float* __restrict__ blk_W2, const float* __restrict__ blk_b2,
                  const float* __restrict__ d_W,    const float* __restrict__ d_b,
                  float* __restrict__ gx, float* __restrict__ gt,
                  float* __restrict__ vout, float* __restrict__ cout)
{
    __shared__ float t1[B_][HIDDEN];
    __shared__ float e [B_][HIDDEN];
    __shared__ float exv[B_][HIDDEN];
    __shared__ float etv[B_][HIDDEN];
    __shared__ float s  [HIDDEN];
    __shared__ float w2d[HIDDEN];

    const int j = threadIdx.x;   // column 0..127

    // t1 = sin(param @ p2e_W1 + b1)
    for (int b = 0; b < B_; ++b) {
        float acc = p2e_b1[j];
        for (int i = 0; i < 16; ++i) acc += param[b * 16 + i] * p2e_W1[i * HIDDEN + j];
        t1[b][j] = __sinf(acc);
    }
    __syncthreads();
    // e = t1 @ p2e_W2 + b2
    for (int b = 0; b < B_; ++b) {
        float acc = p2e_b2[j];
        for (int k = 0; k < HIDDEN; ++k) acc += t1[b][k] * p2e_W2[k * EMB + j];
        e[b][j] = acc;
    }
    __syncthreads();
    // ex / et heads
    for (int b = 0; b < B_; ++b) {
        float ax = e2ex_b[j], at = e2et_b[j];
        for (int k = 0; k < EMB; ++k) {
            float ev = e[b][k];
            ax += ev * e2ex_W[k * EMB + j];
            at += ev * e2et_W[k * EMB + j];
        }
        exv[b][j] = ax; etv[b][j] = at;
    }
    __syncthreads();
    // FiLM gains: gx = ex@px_We + px_be ; gt = et@zt_We + zt_be
    for (int b = 0; b < B_; ++b) {
        float ax = px_be[j], at = zt_be[j];
        for (int k = 0; k < EMB; ++k) {
            ax += exv[b][k] * px_We[k * HIDDEN + j];
            at += etv[b][k] * zt_We[k * HIDDEN + j];
        }
        gx[b * HIDDEN + j] = ax;
        gt[b * HIDDEN + j] = at;
    }
    // s = sin(h0 @ blk_W1 + b1) (single row); w2d = blk_W2 @ d_W
    {
        float acc = blk_b1[j];
        for (int k = 0; k < HIDDEN; ++k) acc += h0[k] * blk_W1[k * HIDDEN + j];
        s[j] = __sinf(acc);
        float w = 0.f;
        for (int k = 0; k < HIDDEN; ++k) w += blk_W2[j * HIDDEN + k] * d_W[k];
        w2d[j] = w;
    }
    __syncthreads();
    // v[d] = sum_j Wc[d,j]*s[j]*w2d[j]
    if (j < 32) {
        float acc = 0.f;
        for (int k = 0; k < HIDDEN; ++k) acc += blk_Wc[j * HIDDEN + k] * s[k] * w2d[k];
        vout[j] = acc;
    }
    // c = dot(bc*s, w2d) + dot(b2, dW) + dot(h0, dW) + d_b
    if (j == 0) {
        float acc = d_b[0];
        for (int k = 0; k < HIDDEN; ++k)
            acc += blk_bc[k] * s[k] * w2d[k] + blk_b2[k] * d_W[k] + h0[k] * d_W[k];
        cout[0] = acc;
    }
}

// ---------------- Stage 1: FiLM'd sin features -----------------------------
// out[b, r, h] = sin(coord[b,r] * Wx[h] + bx[h]) * g[b,h]
__global__ __launch_bounds__(256)
void feat_kernel(const float* __restrict__ coord,
                 const float* __restrict__ Wx, const float* __restrict__ bx,
                 const float* __restrict__ g,
                 float* __restrict__ out, int rows, int total)
{
    int idx = blockIdx.x * 256 + threadIdx.x;
    if (idx >= total) return;
    int h = idx & (HIDDEN - 1);
    int r = (idx >> 7) % rows;
    int b = idx / (rows * HIDDEN);
    float cv = coord[b * rows + r];
    out[idx] = __sinf(cv * Wx[h] + bx[h]) * g[b * HIDDEN + h];
}

// ---------------- Stage 2: feature @ Wo GEMM via fp32 WMMA ----------------
// out[b,m,n]  = mult * (A[b,m,:] @ W[:,n] + bias[n])      (zt / px outputs)
// out2[b,m,n] = out[b,m,n] * vscale[n>>5]                 (pxv, optional)
// One wave computes one 16x16 tile. K=128, stepped by 4 (V_WMMA_F32_16X16X4_F32).
__global__ __launch_bounds__(32)
void gemm_feat_wo(const float* __restrict__ A,     // (B, M, 128)
                  const float* __restrict__ W,     // (128, 1024) row-major
                  const float* __restrict__ bias,  // (1024)
                  const float* __restrict__ mult,  // scalar
                  float* __restrict__ out,         // (B, M, 1024)
                  float* __restrict__ out2,        // optional
                  const float* __restrict__ vscale,// (32)
                  int M)
{
    const int lane = threadIdx.x;
    const int tm = blockIdx.x * 16;
    const int tn = blockIdx.y * 16;
    const int b  = blockIdx.z;
    const int half = lane >> 4;      // selects K+{0,2} per ISA A/B layout
    const int l16  = lane & 15;

    const float* Arow = A + ((size_t)b * M + (tm + l16)) * HIDDEN;
    const float* Wcol = W + tn + l16;

    v8f acc = {};
    for (int k = 0; k < HIDDEN; k += 4) {
        const int ka = k + half * 2;
        v2f av, bv;
        av.x = Arow[ka];
        av.y = Arow[ka + 1];
        bv.x = Wcol[(size_t)ka * DHL];
        bv.y = Wcol[(size_t)(ka + 1) * DHL];
        acc = __builtin_amdgcn_wmma_f32_16x16x4_f32(
            false, av, false, bv, (short)0, acc, false, false);
    }

    const float m  = mult[0];
    const float bn = bias[tn + l16];
    const float vs = out2 ? vscale[(tn + l16) >> 5] : 0.f;
    const size_t base = ((size_t)b * M + tm) * DHL + tn + l16;
#pragma unroll
    for (int r = 0; r < 8; ++r) {
        const int row = r + half * 8;          // C/D layout: lanes16-31 -> M+8
        const float val = m * (acc[r] + bn);
        out[base + (size_t)row * DHL] = val;
        if (out2) out2[base + (size_t)row * DHL] = val * vs;
    }
}

// ---------------- Stage 3: u = (1/32) zt @ pxv^T + c via fp32 WMMA ---------
// U[b,t,x] = (1/32) * sum_k ZT[b,t,k] * PXV[b,x,k] + c.  K=1024 by 4.
__global__ __launch_bounds__(32)
void gemm_u(const float* __restrict__ ZT,   // (B, 256, 1024)
            const float* __restrict__ PXV,  // (B, 128, 1024)
            const float* __restrict__ cptr,
            float* __restrict__ U)          // (B, 256, 128)
{
    const int lane = threadIdx.x;
    const int tm = blockIdx.x * 16;   // t tile
    const int tn = blockIdx.y * 16;   // x tile
    const int b  = blockIdx.z;
    const int half = lane >> 4;
    const int l16  = lane & 15;

    const float* Arow = ZT  + ((size_t)b * T_ + tm + l16) * DHL;
    const float* Brow = PXV + ((size_t)b * X_ + tn + l16) * DHL;

    v8f acc = {};
    for (int k = 0; k < DHL; k += 4) {
        const int ka = k + half * 2;
        v2f av, bv;
        av.x = Arow[ka];
        av.y = Arow[ka + 1];
        bv.x = Brow[ka];          // B[k][n] = PXV[b, n, k] (row-major-T feed)
        bv.y = Brow[ka + 1];
        acc = __builtin_amdgcn_wmma_f32_16x16x4_f32(
            false, av, false, bv, (short)0, acc, false, false);
    }

    const float c = cptr[0];
    const size_t base = ((size_t)b * T_ + tm) * X_ + tn + l16;
#pragma unroll
    for (int r = 0; r < 8; ++r) {
        const int row = r + half * 8;
        U[base + (size_t)row * X_] = acc[r] * (1.0f / 32.0f) + c;
    }
}

// ---------------------------------------------------------------------------
extern "C" void kernel_launch(void* const* d_in, const int* in_sizes, int n_in,
                              void* d_out, int out_size, void* d_ws, size_t ws_size,
                              hipStream_t stream) {
    const float* x      = (const float*)d_in[0];
    const float* t      = (const float*)d_in[1];
    const float* param  = (const float*)d_in[2];
    const float* p2e_W1 = (const float*)d_in[3];
    const float* p2e_b1 = (const float*)d_in[4];
    const float* p2e_W2 = (const float*)d_in[5];
    const float* p2e_b2 = (const float*)d_in[6];
    const float* e2ex_W = (const float*)d_in[7];
    const float* e2ex_b = (const float*)d_in[8];
    const float* e2et_W = (const float*)d_in[9];
    const float* e2et_b = (const float*)d_in[10];
    const float* px_Wx  = (const float*)d_in[11];
    const float* px_bx  = (const float*)d_in[12];
    const float* px_We  = (const float*)d_in[13];
    const float* px_be  = (const float*)d_in[14];
    const float* px_Wo  = (const float*)d_in[15];
    const float* px_bo  = (const float*)d_in[16];
    const float* px_mult= (const float*)d_in[17];
    const float* zt_Wx  = (const float*)d_in[18];
    const float* zt_bx  = (const float*)d_in[19];
    const float* zt_We  = (const float*)d_in[20];
    const float* zt_be  = (const float*)d_in[21];
    const float* zt_Wo  = (const float*)d_in[22];
    const float* zt_bo  = (const float*)d_in[23];
    const float* zt_mult= (const float*)d_in[24];
    const float* h0     = (const float*)d_in[25];
    const float* blk_W1 = (const float*)d_in[26];
    const float* blk_b1 = (const float*)d_in[27];
    const float* blk_Wc = (const float*)d_in[28];
    const float* blk_bc = (const float*)d_in[29];
    const float* blk_W2 = (const float*)d_in[30];
    const float* blk_b2 = (const float*)d_in[31];
    const float* d_W    = (const float*)d_in[32];
    const float* d_b    = (const float*)d_in[33];

    float* out = (float*)d_out;
    float* U   = out;                        // (B,T,X)      131072
    float* ZT  = out + 131072;               // (B,T,1024) 1 048 576
    float* PX  = out + 131072 + 1048576;     // (B,X,1024)   524 288

    float* ws   = (float*)d_ws;
    float* gx   = ws + 0;        //   512
    float* gt   = ws + 512;      //   512
    float* vsc  = ws + 1024;     //    32
    float* cval = ws + 1056;     //     1
    float* fx   = ws + 2048;     //  65536 (B,X,128)
    float* ft   = ws + 67584;    // 131072 (B,T,128)
    float* pxv  = ws + 198656;   // 524288 (B,X,1024)

    // Stage 0: param embeddings + folded BlockMod constants
    setup_kernel<<<1, 128, 0, stream>>>(
        param, p2e_W1, p2e_b1, p2e_W2, p2e_b2,
        e2ex_W, e2ex_b, e2et_W, e2et_b,
        px_We, px_be, zt_We, zt_be,
        h0, blk_W1, blk_b1, blk_Wc, blk_bc, blk_W2, blk_b2,
        d_W, d_b, gx, gt, vsc, cval);

    // Stage 1: sin features
    {
        int total_fx = B_ * X_ * HIDDEN;     // 65536
        feat_kernel<<<(total_fx + 255) / 256, 256, 0, stream>>>(
            x, px_Wx, px_bx, gx, fx, X_, total_fx);
        int total_ft = B_ * T_ * HIDDEN;     // 131072
        feat_kernel<<<(total_ft + 255) / 256, 256, 0, stream>>>(
            t, zt_Wx, zt_bx, gt, ft, T_, total_ft);
    }

    // Stage 2: zt = zt_mult*(ft @ zt_Wo + bo) -> d_out ; px likewise + pxv
    {
        dim3 gzt(T_ / 16, DHL / 16, B_);     // (16,64,4)
        gemm_feat_wo<<<gzt, 32, 0, stream>>>(ft, zt_Wo, zt_bo, zt_mult,
                                             ZT, nullptr, nullptr, T_);
        dim3 gpx(X_ / 16, DHL / 16, B_);     // (8,64,4)
        gemm_feat_wo<<<gpx, 32, 0, stream>>>(fx, px_Wo, px_bo, px_mult,
                                             PX, pxv, vsc, X_);
    }

    // Stage 3: u = (1/32) zt @ pxv^T + c
    {
        dim3 gu(T_ / 16, X_ / 16, B_);       // (16,8,4)
        gemm_u<<<gu, 32, 0, stream>>>(ZT, pxv, cval, U);
    }
}